// ConvCapsuleLayer_24678882082901
// MI455X (gfx1250) — compile-verified
//
#include <hip/hip_runtime.h>
#include <hip/hip_bf16.h>
#include <math.h>

typedef __attribute__((ext_vector_type(2))) float v2f;
typedef __attribute__((ext_vector_type(8))) float v8f;

#define IC 8
#define BATCH 16
#define NC 8
#define NA 16
#define COUT 128            // NC*NA
#define CIN 16
#define KS 5
#define WT_ELEMS (KS*CIN*KS*COUT)        // [kh][ci][kw][co] = 51200
#define WSLAB_ELEMS (CIN*KS*COUT)        // per-kh slice = 10240
#define VOTES_OFF_BYTES (size_t)(1u<<20)

// ---------------- Weight transpose: W[co][ci][kh][kw] -> Wt[kh][ci][kw][co] ----------------
__global__ void wt_prep_kernel(const float* __restrict__ W, float* __restrict__ Wt) {
  int tid = blockIdx.x * blockDim.x + threadIdx.x;
  if (tid >= WT_ELEMS) return;
  int co = tid & 127;
  int r  = tid >> 7;
  int kw = r % 5; r /= 5;
  int ci = r & 15;
  int kh = r >> 4;
  Wt[tid] = W[((co*CIN + ci)*KS + kh)*KS + kw];
}

// ---------------- Implicit-GEMM conv via V_WMMA_F32_16X16X4_F32 ----------------
// Block = 256 threads (8 waves). Block (n, rblk) computes conv[n, 0..127, 2r..2r+1, 0..63]
// and scatters straight into votes[b][h][w][i][cc][nn] layout.
__launch_bounds__(256)
__global__ void conv_wmma_kernel(const float* __restrict__ x,
                                 const float* __restrict__ Wt,
                                 float* __restrict__ votes) {
  __shared__ float inSlab[6*16*68];      // [srow 0..5][ci 0..15][w padded: 2 + 64 + 2]
  __shared__ float wSlab[WSLAB_ELEMS];   // [ci][kw][co]

  const int tid  = threadIdx.x;
  const int wid  = tid >> 5;
  const int lane = tid & 31;
  const int bid  = blockIdx.x;
  const int n    = bid >> 5;     // conv image 0..127
  const int rblk = bid & 31;     // output row-pair: rows 2r, 2r+1

  const int ix = n >> 4;         // input-capsule index for reading x  (n = i*16 + b)
  const int bx = n & 15;

  // zero the padded slab (covers left/right pad cols and out-of-range rows)
  for (int idx = tid; idx < 6*16*68; idx += 256) inSlab[idx] = 0.0f;
  __syncthreads();
  // fill valid interior: input rows 2r-2 .. 2r+3 of the scrambled xr view of x
  for (int idx = tid; idx < 6*16*64; idx += 256) {
    int srow = idx >> 10;
    int ci   = (idx >> 6) & 15;
    int w    = idx & 63;
    int row  = 2*rblk - 2 + srow;
    if (row >= 0 && row < 64) {
      int h  = 4*ci + (row >> 4);
      int wc = 4*(row & 15) + (w >> 4);
      int a  = w & 15;
      inSlab[(srow*16 + ci)*68 + 2 + w] =
          x[(((bx*64 + h)*64 + wc)*8 + ix)*16 + a];
    }
  }

  v8f acc[8] = {};                         // 8 N-tiles of 16 positions (2 rows x 64 cols)
  const int co  = 16*wid + (lane & 15);    // this wave's A rows: co = 16*wid .. +15
  const int kph = (lane >> 4) << 1;        // K half: lanes 0-15 -> K=0,1; lanes 16-31 -> K=2,3
  const int j   = lane & 15;               // N index within tile

  for (int kh = 0; kh < 5; ++kh) {
    __syncthreads();                       // prior-kh compute done before wSlab overwrite
    for (int idx = tid; idx < WSLAB_ELEMS; idx += 256)
      wSlab[idx] = Wt[kh*WSLAB_ELEMS + idx];
    __syncthreads();
    for (int kw = 0; kw < 5; ++kw) {
      #pragma unroll
      for (int k4 = 0; k4 < 4; ++k4) {     // K = 16 input channels in 4x WMMA-K4 steps
        int ciA = 4*k4 + kph;
        v2f a;
        a.x = wSlab[((ciA    )*5 + kw)*128 + co];
        a.y = wSlab[((ciA + 1)*5 + kw)*128 + co];
        #pragma unroll
        for (int g = 0; g < 8; ++g) {
          int srow = (g >> 2) + kh;                // slab row = h2off + kh
          int wj   = 16*(g & 3) + j + kw;          // padded col = (w2 + kw - 2) + 2
          v2f bf;
          bf.x = inSlab[(srow*16 + ciA    )*68 + wj];
          bf.y = inSlab[(srow*16 + ciA + 1)*68 + wj];
          acc[g] = __builtin_amdgcn_wmma_f32_16x16x4_f32(
              false, a, false, bf, (short)0, acc[g], false, false);
        }
      }
    }
  }

  // Epilogue: conv[n,c,h2,w2] -> votes[b=n/8][h=c/2][w=rblk+32*(c&1)][i=n%8][cc][nn]
  const int b_v = n >> 3;
  const int i_v = n & 7;
  #pragma unroll
  for (int g = 0; g < 8; ++g) {
    int cc = (g & 3) + 4*(g >> 2);               // (w2>>4) + 4*(h2&1)
    #pragma unroll
    for (int v = 0; v < 8; ++v) {
      int c  = 16*wid + v + 8*(lane >> 4);       // D-matrix M = v + 8*(lane/16)
      int h  = c >> 1;
      int w  = rblk + 32*(c & 1);
      int nn = lane & 15;
      votes[((((b_v*64 + h)*64 + (long)w)*8 + i_v)*8 + cc)*16 + nn] = acc[g][v];
    }
  }
}

// ---------------- Dynamic routing: one 128-thread block per (b,h,w) position ----------------
__launch_bounds__(128)
__global__ void routing_kernel(const float* __restrict__ votes,
                               const float* __restrict__ bias,
                               float* __restrict__ out) {
  __shared__ float logits_s[IC][NC];
  __shared__ float route_s[IC][NC];
  const int t  = threadIdx.x;        // t = cc*16 + nn
  const int cc = t >> 4;
  const int nn = t & 15;
  const long pos = blockIdx.x;       // (b*64+h)*64+w
  const float* vbase = votes + pos * 1024;

  float v[IC];
  #pragma unroll
  for (int i = 0; i < IC; ++i) v[i] = vbase[i*128 + t];
  const float bz = bias[t];
  if (t < IC*NC) ((float*)logits_s)[t] = 0.0f;

  float act = 0.0f;
  for (int it = 0; it < 3; ++it) {
    __syncthreads();
    if (t < IC*NC) {                          // softmax over nc per input capsule
      int i = t >> 3;
      int c = t & 7;
      float m = -1e30f;
      #pragma unroll
      for (int c2 = 0; c2 < NC; ++c2) m = fmaxf(m, logits_s[i][c2]);
      float s = 0.0f;
      #pragma unroll
      for (int c2 = 0; c2 < NC; ++c2) s += expf(logits_s[i][c2] - m);
      route_s[i][c] = expf(logits_s[i][c] - m) / s;
    }
    __syncthreads();
    float p = bz;
    #pragma unroll
    for (int i = 0; i < IC; ++i) p += route_s[i][cc] * v[i];
    // squash: reduce p^2 over the 16 atoms (16-lane aligned groups in wave32)
    float sq = p * p;
    sq += __shfl_xor(sq, 1, 32);
    sq += __shfl_xor(sq, 2, 32);
    sq += __shfl_xor(sq, 4, 32);
    sq += __shfl_xor(sq, 8, 32);
    act = p * sq / ((1.0f + sq) * sqrtf(sq + 1e-12f));
    if (it < 2) {                             // agreement update (unused after last iter)
      float s_i[IC];
      #pragma unroll
      for (int i = 0; i < IC; ++i) {
        float r = v[i] * act;
        r += __shfl_xor(r, 1, 32);
        r += __shfl_xor(r, 2, 32);
        r += __shfl_xor(r, 4, 32);
        r += __shfl_xor(r, 8, 32);
        s_i[i] = r;
      }
      if (nn == 0) {
        #pragma unroll
        for (int i = 0; i < IC; ++i) logits_s[i][cc] += s_i[i];
      }
    }
  }
  out[pos*128 + t] = act;
}

extern "C" void kernel_launch(void* const* d_in, const int* in_sizes, int n_in,
                              void* d_out, int out_size, void* d_ws, size_t ws_size,
                              hipStream_t stream) {
  const float* x = (const float*)d_in[0];   // (16,64,64,8,16)
  const float* W = (const float*)d_in[1];   // (128,16,5,5)
  const float* b = (const float*)d_in[2];   // (1,1,8,16)
  float* out   = (float*)d_out;             // (16,64,64,8,16)
  float* Wt    = (float*)d_ws;                                   // 200 KB
  float* votes = (float*)((char*)d_ws + VOTES_OFF_BYTES);        // 256 MB

  wt_prep_kernel<<<(WT_ELEMS + 255)/256, 256, 0, stream>>>(W, Wt);
  conv_wmma_kernel<<<128*32, 256, 0, stream>>>(x, Wt, votes);
  routing_kernel<<<BATCH*64*64, 128, 0, stream>>>(votes, b, out);
}